// SRenderY_24361054503560
// MI455X (gfx1250) — compile-verified
//
#include <hip/hip_runtime.h>
#include <hip/hip_bf16.h>

// Problem constants (from reference): B=2, V=5023, F=2048, H=W=256
#define BATCH 2
#define NVERT 5023
#define NFACE 2048
#define HH 256
#define WW 256
#define NPIX (HH * WW)          // 65536
#define BF (BATCH * NFACE)      // 4096

// LDS staging: 4 coefficient planes (float2 each) + constant terms (float4)
#define SMEM_CXY_BYTES (4 * NFACE * 8)           // 65536
#define SMEM_CC_BYTES  (NFACE * 16)              // 32768
#define SMEM_BYTES     (SMEM_CXY_BYTES + SMEM_CC_BYTES)  // 98304 (<=320KB WGP LDS)

typedef __attribute__((ext_vector_type(2))) float v2f;
typedef __attribute__((ext_vector_type(8))) float v8f;

// Types for the async Global->LDS builtin (param type per hipcc diagnostic:
// '__attribute__((vector_size(4*sizeof(int)))) int *').
typedef int v4i_b128 __attribute__((vector_size(16)));
typedef __attribute__((address_space(3))) v4i_b128 lds_v4i_b128;

// Async 16B Global -> LDS copy (ASYNCcnt-tracked). Fallback: load+ds_store.
#if __has_builtin(__builtin_amdgcn_global_load_async_to_lds_b128)
#define HAVE_ASYNC_LDS 1
#define ASYNC_CP16(dst_lds, src_glob)                                          \
    __builtin_amdgcn_global_load_async_to_lds_b128(                            \
        (v4i_b128*)(src_glob), (lds_v4i_b128*)(dst_lds), 0, 0)
#else
#define HAVE_ASYNC_LDS 0
#define ASYNC_CP16(dst_lds, src_glob)                                          \
    (*(float4*)(dst_lds) = *(const float4*)(src_glob))
#endif

static __device__ __forceinline__ void wait_async_then_barrier() {
#if HAVE_ASYNC_LDS
#if __has_builtin(__builtin_amdgcn_s_wait_asynccnt)
    __builtin_amdgcn_s_wait_asynccnt(0);
#else
    asm volatile("s_wait_asynccnt 0x0" ::: "memory");
#endif
#endif
    __syncthreads();
}

// ---------------------------------------------------------------------------
// Kernel 1: per-(batch,face) edge-function plane coefficients.
//   w0(px,py) = c0x*px + c0y*py + c0c      (>=0 inside)
//   w1(px,py) = c1x*px + c1y*py + c1c
//   w2(px,py) = c2x*px + c2y*py + c2c      (== 1 - w0 - w1 as a plane)
//   z (px,py) = czx*px + czy*py + czc
// Degenerate faces (|denom|<=1e-8) get w0/w1 planes == -1 so they never pass
// the inside test, matching the reference's `inside & (|denom|>1e-8)`.
// Plane layout: c_xy[plane][b*NFACE+f], plane 0..3 = w0,w1,w2,z.
// ---------------------------------------------------------------------------
__global__ void prep_faces_kernel(const float* __restrict__ tv,
                                  const int* __restrict__ faces,
                                  float2* __restrict__ c_xy,   // [4][BF]
                                  float4* __restrict__ c_c)    // [BF] = {c0c,c1c,c2c,czc}
{
    int idx = blockIdx.x * blockDim.x + threadIdx.x;
    if (idx >= BF) return;
    int b = idx / NFACE;
    int f = idx - b * NFACE;
    int i0 = faces[3 * f + 0], i1 = faces[3 * f + 1], i2 = faces[3 * f + 2];
    const float* p0 = tv + ((size_t)b * NVERT + i0) * 3;
    const float* p1 = tv + ((size_t)b * NVERT + i1) * 3;
    const float* p2 = tv + ((size_t)b * NVERT + i2) * 3;
    float x0 = p0[0], y0 = p0[1], z0 = p0[2];
    float x1 = p1[0], y1 = p1[1], z1 = p1[2];
    float x2 = p2[0], y2 = p2[1], z2 = p2[2];

    float denom = (y1 - y2) * (x0 - x2) + (x2 - x1) * (y0 - y2);
    bool ok = fabsf(denom) > 1e-8f;
    float inv = ok ? (1.0f / denom) : 0.0f;

    float c0x = (y1 - y2) * inv, c0y = (x2 - x1) * inv;
    float c1x = (y2 - y0) * inv, c1y = (x0 - x2) * inv;
    float c0c = -(c0x * x2 + c0y * y2);
    float c1c = -(c1x * x2 + c1y * y2);
    if (!ok) { c0x = 0.f; c0y = 0.f; c0c = -1.0f; c1x = 0.f; c1y = 0.f; c1c = -1.0f; }

    float c2x = -(c0x + c1x);
    float c2y = -(c0y + c1y);
    float c2c = 1.0f - c0c - c1c;

    float dz0 = z0 - z2, dz1 = z1 - z2;
    float czx = c0x * dz0 + c1x * dz1;
    float czy = c0y * dz0 + c1y * dz1;
    float czc = c0c * dz0 + c1c * dz1 + z2;

    c_xy[0 * BF + idx] = make_float2(c0x, c0y);
    c_xy[1 * BF + idx] = make_float2(c1x, c1y);
    c_xy[2 * BF + idx] = make_float2(c2x, c2y);
    c_xy[3 * BF + idx] = make_float2(czx, czy);
    c_c[idx] = make_float4(c0c, c1c, c2c, czc);
}

// ---------------------------------------------------------------------------
// Kernel 2: per-(batch,vertex) attributes: bilinear texture sample + xyz.
// ---------------------------------------------------------------------------
__global__ void prep_verts_kernel(const float* __restrict__ verts,
                                  const float* __restrict__ tex,  // [B][256][256][3]
                                  const float* __restrict__ uv,   // [NVERT][2]
                                  float* __restrict__ vattr)      // [B][NVERT][6]
{
    int idx = blockIdx.x * blockDim.x + threadIdx.x;
    if (idx >= BATCH * NVERT) return;
    int b = idx / NVERT;
    int v = idx - b * NVERT;

    float u = uv[2 * v + 0] * 255.0f;
    float w = uv[2 * v + 1] * 255.0f;
    float u0f = fminf(fmaxf(floorf(u), 0.0f), 254.0f);
    float v0f = fminf(fmaxf(floorf(w), 0.0f), 254.0f);
    int u0 = (int)u0f, v0 = (int)v0f;
    float fu = u - u0f, fv = w - v0f;

    const float* t   = tex + (size_t)b * NPIX * 3;
    const float* c00 = t + ((size_t)v0 * 256 + u0) * 3;
    const float* c01 = c00 + 3;
    const float* c10 = c00 + 256 * 3;
    const float* c11 = c10 + 3;

    float* o = vattr + ((size_t)b * NVERT + v) * 6;
#pragma unroll
    for (int c = 0; c < 3; ++c) {
        float top = c00[c] * (1.0f - fu) + c01[c] * fu;
        float bot = c10[c] * (1.0f - fu) + c11[c] * fu;
        o[c] = top * (1.0f - fv) + bot * fv;
    }
    const float* vt = verts + ((size_t)b * NVERT + v) * 3;
    o[3] = vt[0]; o[4] = vt[1]; o[5] = vt[2];
}

// ---------------------------------------------------------------------------
// Kernel 3: WMMA rasterizer + shader.
// Per block: async-stage this batch's full coefficient tables (96 KB) into
// LDS, then each wave rasterizes a 16-pixel strip against all 2048 faces:
//   4x V_WMMA_F32_16X16X4_F32 per 16-face group (w0, w1, w2, z planes),
//   inside test = min3 >= 0, strict-> argmax scan + shfl_xor(16) half-combine
// reproduces "lowest face index among max z, z > -1" exactly.
// ---------------------------------------------------------------------------
__global__ __launch_bounds__(256) void raster_kernel(
    const float2* __restrict__ c_xy,    // [4][BF]
    const float4* __restrict__ c_c,     // [BF]
    const float* __restrict__ vattr,    // [B][NVERT][6]
    const int* __restrict__ faces,      // [NFACE][3]
    float* __restrict__ out_render,     // [B][6][H][W]
    float* __restrict__ out_alpha,      // [B][H][W]
    float* __restrict__ out_fid,        // [B][H][W]  (fid as float value)
    float* __restrict__ out_zbuf)       // [B][H][W]
{
    extern __shared__ char smem[];
    float2* s_cxy = (float2*)smem;                       // [4][NFACE]
    float4* s_cc  = (float4*)(smem + SMEM_CXY_BYTES);    // [NFACE]

    const int tid   = threadIdx.x;
    const int lane  = tid & 31;
    const int wave  = tid >> 5;
    const int b     = blockIdx.y;

    // ---- Stage coefficients: Global -> LDS (async, ASYNCcnt-tracked) ----
#pragma unroll
    for (int pl = 0; pl < 4; ++pl) {
        const float2* gsrc = c_xy + (size_t)pl * BF + (size_t)b * NFACE;
        float2* ldst = s_cxy + pl * NFACE;
        for (int i = tid * 2; i < NFACE; i += 256 * 2)   // 16B per copy
            ASYNC_CP16(&ldst[i], &gsrc[i]);
    }
    {
        const float4* gsrc = c_c + (size_t)b * NFACE;
        for (int i = tid; i < NFACE; i += 256)
            ASYNC_CP16(&s_cc[i], &gsrc[i]);
    }
    wait_async_then_barrier();

    // ---- Per-wave 16-pixel strip ----
    const int n     = lane & 15;                          // pixel column in tile
    const int p     = blockIdx.x * 128 + wave * 16 + n;   // linear pixel
    const int xpix  = p & (WW - 1);
    const int ypix  = p >> 8;
    const bool lower = (lane < 16);

    // B matrix (4x16): K0 = px, K1 = py, K2 = 1, K3 = 0
    v2f bmat;
    bmat.x = lower ? (((float)xpix + 0.5f) * (2.0f / WW) - 1.0f) : 1.0f;
    bmat.y = lower ? (((float)ypix + 0.5f) * (2.0f / HH) - 1.0f) : 0.0f;

    float bz = -1.0f, bw0 = 0.0f, bw1 = 0.0f;   // zbuf init -1.0 => z > -1 threshold
    int   bi = -1;

    for (int g = 0; g < NFACE / 16; ++g) {
        const int fg = g * 16 + n;              // this lane's triangle row M = n
        v2f a0, a1, a2, az;
        if (lower) {                            // A[m][K0..K1] for 4 planes
            float2 t0 = s_cxy[0 * NFACE + fg];
            float2 t1 = s_cxy[1 * NFACE + fg];
            float2 t2 = s_cxy[2 * NFACE + fg];
            float2 t3 = s_cxy[3 * NFACE + fg];
            a0.x = t0.x; a0.y = t0.y;
            a1.x = t1.x; a1.y = t1.y;
            a2.x = t2.x; a2.y = t2.y;
            az.x = t3.x; az.y = t3.y;
        } else {                                // A[m][K2..K3]
            float4 cc = s_cc[fg];
            a0.x = cc.x; a0.y = 0.0f;
            a1.x = cc.y; a1.y = 0.0f;
            a2.x = cc.z; a2.y = 0.0f;
            az.x = cc.w; az.y = 0.0f;
        }

        v8f zero = {0.f, 0.f, 0.f, 0.f, 0.f, 0.f, 0.f, 0.f};
        v8f W0 = __builtin_amdgcn_wmma_f32_16x16x4_f32(false, a0, false, bmat,
                                                       (short)0, zero, false, false);
        v8f W1 = __builtin_amdgcn_wmma_f32_16x16x4_f32(false, a1, false, bmat,
                                                       (short)0, zero, false, false);
        v8f W2 = __builtin_amdgcn_wmma_f32_16x16x4_f32(false, a2, false, bmat,
                                                       (short)0, zero, false, false);
        v8f ZZ = __builtin_amdgcn_wmma_f32_16x16x4_f32(false, az, false, bmat,
                                                       (short)0, zero, false, false);

        // Per-half scan over 8 triangles (ascending index, strict >)
        float gz = -INFINITY, gw0 = 0.0f, gw1 = 0.0f;
        int gi = -1;
#pragma unroll
        for (int r = 0; r < 8; ++r) {
            float m = fminf(fminf(W0[r], W1[r]), W2[r]);   // v_min3
            float z = (m >= 0.0f) ? ZZ[r] : -INFINITY;
            if (z > gz) { gz = z; gi = r; gw0 = W0[r]; gw1 = W1[r]; }
        }
        gi = (gi >= 0) ? (g * 16 + (lower ? 0 : 8) + gi) : -1;

        // Combine halves: upper half holds higher face indices -> strict > to win
        float oz  = __shfl_xor(gz, 16, 32);
        int   oi  = __shfl_xor(gi, 16, 32);
        float ow0 = __shfl_xor(gw0, 16, 32);
        float ow1 = __shfl_xor(gw1, 16, 32);
        bool takeOther = lower ? (oz > gz) : !(gz > oz);
        if (takeOther) { gz = oz; gi = oi; gw0 = ow0; gw1 = ow1; }

        // Merge into running best (later groups have higher indices -> strict >)
        if (gz > bz) { bz = gz; bi = gi; bw0 = gw0; bw1 = gw1; }
    }

    if (lower) {
        float a[6] = {0.f, 0.f, 0.f, 0.f, 0.f, 0.f};
        float alpha = 0.0f;
        if (bi >= 0) {
            alpha = 1.0f;
            int i0 = faces[3 * bi + 0], i1 = faces[3 * bi + 1], i2 = faces[3 * bi + 2];
            const float* A0 = vattr + ((size_t)b * NVERT + i0) * 6;
            const float* A1 = vattr + ((size_t)b * NVERT + i1) * 6;
            const float* A2 = vattr + ((size_t)b * NVERT + i2) * 6;
            float w2 = 1.0f - bw0 - bw1;   // reference bary w2 = 1 - w0 - w1
#pragma unroll
            for (int c = 0; c < 6; ++c)
                a[c] = bw0 * A0[c] + bw1 * A1[c] + w2 * A2[c];
        }
        const size_t pb = (size_t)b * NPIX + p;
#pragma unroll
        for (int c = 0; c < 6; ++c)
            out_render[(size_t)b * 6 * NPIX + (size_t)c * NPIX + p] = a[c];
        out_alpha[pb] = alpha;
        out_fid[pb]   = (float)bi;   // int fid emitted as float value in concat output
        out_zbuf[pb]  = bz;
    }
}

// ---------------------------------------------------------------------------
extern "C" void kernel_launch(void* const* d_in, const int* in_sizes, int n_in,
                              void* d_out, int out_size, void* d_ws, size_t ws_size,
                              hipStream_t stream) {
    const float* vertices = (const float*)d_in[0];  // [B][NVERT][3]
    const float* tverts   = (const float*)d_in[1];  // [B][NVERT][3]
    const float* textures = (const float*)d_in[2];  // [B][256][256][3]
    const float* uvcoords = (const float*)d_in[3];  // [NVERT][2]
    const int*   faces    = (const int*)d_in[4];    // [NFACE][3]

    // Workspace partition
    char*   ws    = (char*)d_ws;
    float2* c_xy  = (float2*)ws;                          // 4*BF*8   = 131072 B
    float4* c_c   = (float4*)(ws + 131072);               // BF*16    =  65536 B
    float*  vattr = (float*)(ws + 131072 + 65536);        // B*NVERT*6*4 = 241104 B

    // Output partition (tuple concatenated flat, return order)
    float* out        = (float*)d_out;
    float* out_render = out;                              // B*6*H*W = 786432
    float* out_alpha  = out + (size_t)BATCH * 6 * NPIX;   // B*1*H*W = 131072
    float* out_fid    = out_alpha + (size_t)BATCH * NPIX;
    float* out_zbuf   = out_fid + (size_t)BATCH * NPIX;

    // Allow 96KB dynamic LDS (WGP has 320KB).
    (void)hipFuncSetAttribute((const void*)raster_kernel,
                              hipFuncAttributeMaxDynamicSharedMemorySize,
                              SMEM_BYTES);

    prep_faces_kernel<<<(BF + 255) / 256, 256, 0, stream>>>(tverts, faces, c_xy, c_c);
    prep_verts_kernel<<<(BATCH * NVERT + 255) / 256, 256, 0, stream>>>(
        vertices, textures, uvcoords, vattr);

    dim3 grid(NPIX / 128, BATCH);  // 8 waves/block * 16 px/wave = 128 px/block
    raster_kernel<<<grid, 256, SMEM_BYTES, stream>>>(c_xy, c_c, vattr, faces,
                                                     out_render, out_alpha,
                                                     out_fid, out_zbuf);
}